// PointPillarFeatureNet_40235253629487
// MI455X (gfx1250) — compile-verified
//
#include <hip/hip_runtime.h>
#include <math.h>

typedef float v2f __attribute__((ext_vector_type(2)));
typedef float v4f __attribute__((ext_vector_type(4)));
typedef float v8f __attribute__((ext_vector_type(8)));

#define PFN_VX    0.2f
#define PFN_VY    0.2f
#define PFN_XOFF  (0.1f)            /* VX/2 + 0.0   */
#define PFN_YOFF  (0.1f - 25.6f)    /* VY/2 - 25.6  */
#define PFN_EPS   1e-3f

// One wave (32 lanes) processes one pillar (32 points).
// GEMM: [32 x 12] @ [12 x 64] via V_WMMA_F32_16X16X4_F32 (fp32-exact).
// K layout (12): [x,y,z,w, x-mx,y-my,z-mz, x-cx,y-cy, 1, 0, 0]
// W' layout   : W'[k][c] = W[k][c]*inv[c] (k<9); W'[9][c] = beta-mean*inv; rows 10,11 = 0.
__global__ __launch_bounds__(256) void pfn_wmma_kernel(
    const float* __restrict__ voxels,      // [N,32,4]
    const int*   __restrict__ num_points,  // [N]
    const int*   __restrict__ coords,      // [N,3]
    const float* __restrict__ Wm,          // [9,64]
    const float* __restrict__ gammav,      // [64]
    const float* __restrict__ betav,       // [64]
    const float* __restrict__ rmean,       // [64]
    const float* __restrict__ rvar,        // [64]
    float*       __restrict__ out,         // [N,64]
    int N)
{
    __shared__ float s_wt[64 * 12];        // W' transposed: [col 64][k 12]
    __shared__ float s_feat[8 * 32 * 12];  // per-wave point features

    const int tid = threadIdx.x;

    // ---- Build folded weights W' cooperatively (768 entries / 256 threads) ----
    for (int i = tid; i < 64 * 12; i += 256) {
        const int c = i / 12;
        const int k = i - c * 12;
        const float inv = gammav[c] / sqrtf(rvar[c] + PFN_EPS);
        float v;
        if (k < 9)       v = Wm[k * 64 + c] * inv;
        else if (k == 9) v = betav[c] - rmean[c] * inv;
        else             v = 0.0f;
        s_wt[c * 12 + k] = v;
    }
    __syncthreads();

    const int lane = tid & 31;
    const int wave = tid >> 5;
    const int hi   = lane >> 4;      // lane-half selects upper K pair
    const int l15  = lane & 15;

    // ---- B fragments (pillar-invariant, kept in registers) ----
    // B 4x16 f32 layout: lanes 0-15 hold K={4ks,4ks+1}, lanes 16-31 hold K={4ks+2,4ks+3}.
    v2f bfrag[3][4];
#pragma unroll
    for (int ks = 0; ks < 3; ++ks)
#pragma unroll
        for (int nt = 0; nt < 4; ++nt) {
            const int col = nt * 16 + l15;
            bfrag[ks][nt] = *(const v2f*)&s_wt[col * 12 + ks * 4 + hi * 2];
        }

    // Clamp (not branch) so EXEC stays all-ones through WMMA region.
    long pillar = (long)blockIdx.x * 8 + wave;
    if (pillar >= N) pillar = N - 1;

    // ---- Per-lane point: load + features ----
    const v4f  p  = *(const v4f*)&voxels[(pillar * 32 + lane) * 4];
    const int  np = num_points[pillar];
    const float valid = (lane < np) ? 1.0f : 0.0f;

    // masked mean of xyz over the wave (butterfly reduce, wave32)
    float sx = p.x * valid, sy = p.y * valid, sz = p.z * valid;
#pragma unroll
    for (int m = 16; m >= 1; m >>= 1) {
        sx += __shfl_xor(sx, m, 32);
        sy += __shfl_xor(sy, m, 32);
        sz += __shfl_xor(sz, m, 32);
    }
    const float rnp = 1.0f / (float)np;
    const float mx = sx * rnp, my = sy * rnp, mz = sz * rnp;

    const float cx = (float)coords[pillar * 3 + 0] * PFN_VX + PFN_XOFF;
    const float cy = (float)coords[pillar * 3 + 1] * PFN_VY + PFN_YOFF;

    float* frow = &s_feat[wave * 384 + lane * 12];
    v4f f0 = { p.x * valid, p.y * valid, p.z * valid, p.w * valid };
    v4f f1 = { (p.x - mx) * valid, (p.y - my) * valid,
               (p.z - mz) * valid, (p.x - cx) * valid };
    v4f f2 = { (p.y - cy) * valid, 1.0f, 0.0f, 0.0f };
    *(v4f*)&frow[0] = f0;
    *(v4f*)&frow[4] = f1;
    *(v4f*)&frow[8] = f2;
    // Same-wave LDS ops are in-order (DScnt): A-fragment loads below see stores.

    // ---- 24x V_WMMA_F32_16X16X4_F32 : 2 M-tiles x 4 N-tiles x 3 K-steps ----
    v8f acc[2][4] = {};
    const float* fb = &s_feat[wave * 384];
#pragma unroll
    for (int mt = 0; mt < 2; ++mt) {
#pragma unroll
        for (int ks = 0; ks < 3; ++ks) {
            const int row = mt * 16 + l15;
            // A 16x4 f32 layout: lanes 0-15 K={4ks,4ks+1}; lanes 16-31 K={4ks+2,4ks+3}.
            const v2f a = *(const v2f*)&fb[row * 12 + ks * 4 + hi * 2];
#pragma unroll
            for (int nt = 0; nt < 4; ++nt) {
                acc[mt][nt] = __builtin_amdgcn_wmma_f32_16x16x4_f32(
                    false, a, false, bfrag[ks][nt],
                    (short)0, acc[mt][nt], false, false);
            }
        }
    }

    // ---- ReLU (seed 0) + max over 16 rows/tile + combine lane halves + tiles ----
#pragma unroll
    for (int nt = 0; nt < 4; ++nt) {
        float r = 0.0f;  // implicit ReLU
#pragma unroll
        for (int i = 0; i < 8; ++i)
            r = fmaxf(r, fmaxf(acc[0][nt][i], acc[1][nt][i]));
        // lanes 0-15 hold max over rows 0-7, lanes 16-31 over rows 8-15: combine.
        r = fmaxf(r, __shfl_xor(r, 16, 32));
        if (lane < 16)
            out[pillar * 64 + nt * 16 + l15] = r;
    }
}

extern "C" void kernel_launch(void* const* d_in, const int* in_sizes, int n_in,
                              void* d_out, int out_size, void* d_ws, size_t ws_size,
                              hipStream_t stream) {
    const float* voxels     = (const float*)d_in[0];
    const int*   num_points = (const int*)  d_in[1];
    const int*   coords     = (const int*)  d_in[2];
    const float* Wm         = (const float*)d_in[3];
    const float* gammav     = (const float*)d_in[4];
    const float* betav      = (const float*)d_in[5];
    const float* rmean      = (const float*)d_in[6];
    const float* rvar       = (const float*)d_in[7];
    float* out = (float*)d_out;

    const int N = in_sizes[1];          // num_points has N elements
    const int blocks = (N + 7) / 8;     // 8 pillars (waves) per 256-thread block
    hipLaunchKernelGGL(pfn_wmma_kernel, dim3(blocks), dim3(256), 0, stream,
                       voxels, num_points, coords, Wm, gammav, betav, rmean, rvar,
                       out, N);
}